// SoftCrossEntropyLoss_25409026523315
// MI455X (gfx1250) — compile-verified
//
#include <hip/hip_runtime.h>
#include <math.h>

// SoftCrossEntropyLoss: mean_n [ lse_n * sum_k(t) - sum_k(t*x) ],
// lse_n = max_k x + log(sum_k exp(x - max)).
// N = 131072 rows, K = 512 cols, f32 in / f32 scalar out.
// Memory-bound: 512 MB streamed once -> NT loads, coalesced b128, wave/row.

typedef float vf4 __attribute__((ext_vector_type(4)));
typedef __attribute__((ext_vector_type(2))) float v2f;
typedef __attribute__((ext_vector_type(8))) float v8f;

#define KDIM 512
#define ROWS_PER_BLOCK 8   // 8 waves of 32 per block

// Wave32 sum reduction via V_WMMA_F32_16X16X4_F32.
// A-matrix 16x4 layout: A[m][0]=V0@lane m, A[m][1]=V1@lane m,
//                       A[m][2]=V0@lane m+16, A[m][3]=V1@lane m+16.
// With a0 = v, a1 = 0, B = ones: D[m][*] = v[m] + v[m+16].
// C/D layout: VGPR r holds M=r (lanes 0-15) and M=r+8 (lanes 16-31), so
// sum(c[0..7]) gives half the rows; shfl_xor(16) completes the reduction.
__device__ __forceinline__ float wave_sum_wmma(float v) {
  v2f a; a[0] = v;    a[1] = 0.0f;
  v2f b; b[0] = 1.0f; b[1] = 1.0f;
  v8f c = {};
  c = __builtin_amdgcn_wmma_f32_16x16x4_f32(
      /*neg_a=*/false, a, /*neg_b=*/false, b,
      /*c_mod=*/(short)0, c, /*reuse_a=*/false, /*reuse_b=*/false);
  float s = ((c[0] + c[1]) + (c[2] + c[3])) + ((c[4] + c[5]) + (c[6] + c[7]));
  s += __shfl_xor(s, 16, 32);
  return s;
}

__global__ __launch_bounds__(ROWS_PER_BLOCK * 32)
void sce_rows_kernel(const float* __restrict__ x,
                     const float* __restrict__ t,
                     float* __restrict__ partials) {
  const int lane = threadIdx.x & 31;
  const int wave = threadIdx.x >> 5;
  const size_t row = (size_t)blockIdx.x * ROWS_PER_BLOCK + wave;

  const vf4* __restrict__ xr = (const vf4*)(x + row * KDIM);
  const vf4* __restrict__ tr = (const vf4*)(t + row * KDIM);

  // 4 x b128 per operand: each load covers 512 contiguous bytes per wave.
  vf4 xv[4], tv[4];
#pragma unroll
  for (int i = 0; i < 4; ++i) {
    xv[i] = __builtin_nontemporal_load(&xr[lane + 32 * i]);
    tv[i] = __builtin_nontemporal_load(&tr[lane + 32 * i]);
  }

  // Lane-local max over 16 elements.
  float m = -INFINITY;
#pragma unroll
  for (int i = 0; i < 4; ++i) {
    m = fmaxf(m, fmaxf(fmaxf(xv[i].x, xv[i].y), fmaxf(xv[i].z, xv[i].w)));
  }
  // Wave-wide max: 5-step butterfly.
#pragma unroll
  for (int off = 16; off > 0; off >>= 1) {
    m = fmaxf(m, __shfl_xor(m, off, 32));
  }

  // Lane-local partials.
  float se = 0.0f, dot = 0.0f, ts = 0.0f;
#pragma unroll
  for (int i = 0; i < 4; ++i) {
    se  += __expf(xv[i].x - m) + __expf(xv[i].y - m)
         + __expf(xv[i].z - m) + __expf(xv[i].w - m);
    dot += tv[i].x * xv[i].x + tv[i].y * xv[i].y
         + tv[i].z * xv[i].z + tv[i].w * xv[i].w;
    ts  += (tv[i].x + tv[i].y) + (tv[i].z + tv[i].w);
  }

  // Wave-wide sums via WMMA (whole wave convergent here: EXEC all ones).
  se  = wave_sum_wmma(se);
  dot = wave_sum_wmma(dot);
  ts  = wave_sum_wmma(ts);

  const float loss = (m + __logf(se)) * ts - dot;

  __shared__ float wl[ROWS_PER_BLOCK];
  if (lane == 0) wl[wave] = loss;
  __syncthreads();
  if (threadIdx.x == 0) {
    float s = 0.0f;
#pragma unroll
    for (int i = 0; i < ROWS_PER_BLOCK; ++i) s += wl[i];
    partials[blockIdx.x] = s;  // fixed order per block -> deterministic
  }
}

__global__ __launch_bounds__(1024)
void sce_final_kernel(const float* __restrict__ partials, int n_part,
                      float inv_n, float* __restrict__ out) {
  __shared__ float sm[1024];
  float s = 0.0f;
  // Fixed-stride strided sum: deterministic order per thread.
  for (int i = threadIdx.x; i < n_part; i += 1024) s += partials[i];
  sm[threadIdx.x] = s;
  __syncthreads();
#pragma unroll
  for (int step = 512; step > 0; step >>= 1) {
    if ((int)threadIdx.x < step) sm[threadIdx.x] += sm[threadIdx.x + step];
    __syncthreads();
  }
  if (threadIdx.x == 0) out[0] = sm[0] * inv_n;
}

extern "C" void kernel_launch(void* const* d_in, const int* in_sizes, int n_in,
                              void* d_out, int out_size, void* d_ws, size_t ws_size,
                              hipStream_t stream) {
  const float* x = (const float*)d_in[0];   // input  [N, 512] f32
  const float* t = (const float*)d_in[1];   // target [N, 512] f32
  float* out = (float*)d_out;               // scalar f32

  const int n_rows  = in_sizes[0] / KDIM;                 // 131072
  const int nblocks = n_rows / ROWS_PER_BLOCK;            // 16384
  float* partials = (float*)d_ws;                         // 64 KB scratch

  sce_rows_kernel<<<nblocks, ROWS_PER_BLOCK * 32, 0, stream>>>(x, t, partials);
  sce_final_kernel<<<1, 1024, 0, stream>>>(partials, nblocks,
                                           1.0f / (float)n_rows, out);
}